// PointMLPFeaturePropagation_81870666596672
// MI455X (gfx1250) — compile-verified
//
#include <hip/hip_runtime.h>
#include <stdint.h>

#define EPS_BN 1e-5f
#define EPS_W  1e-8f

typedef float v2f __attribute__((ext_vector_type(2)));
typedef float v8f __attribute__((ext_vector_type(8)));

// ---------------------------------------------------------------------------
// CDNA5 async copy: global -> LDS, 16B per lane, tracked by ASYNCcnt.
// Builtins don't cover these; inline asm per CDNA5 ISA.
//   global_load_async_to_lds_b128  vdst(LDS byte addr), vaddr(u32 offset), saddr
// ---------------------------------------------------------------------------
__device__ __forceinline__ void async_load_b128(uint32_t lds_byte_addr,
                                                const void* sbase,
                                                uint32_t voff_bytes) {
    asm volatile("global_load_async_to_lds_b128 %0, %1, %2 offset:0"
                 :: "v"(lds_byte_addr), "v"(voff_bytes), "s"(sbase)
                 : "memory");
}
__device__ __forceinline__ void wait_async0() {
    asm volatile("s_wait_asynccnt 0x0" ::: "memory");
}
__device__ __forceinline__ uint32_t lds_addr_of(const void* p) {
    return (uint32_t)(uintptr_t)p;   // addrspace(3) ptr -> byte offset
}

// ---------------------------------------------------------------------------
// Kernel 1: 3-NN + inverse-distance weights. grid=(N/256,B), block=256.
// ---------------------------------------------------------------------------
__global__ __launch_bounds__(256)
void knn3_kernel(const float* __restrict__ xyz1,   // (B,N,3)
                 const float* __restrict__ xyz2,   // (B,S,3)
                 float* __restrict__ w_ws,         // (B,N,3)
                 int*   __restrict__ idx_ws,       // (B,N,3)
                 int N, int S)
{
    __shared__ float sx[1024], sy[1024], sz[1024];
    const int b = blockIdx.y;
    const int n = blockIdx.x * 256 + threadIdx.x;

    for (int s = threadIdx.x; s < S; s += 256) {
        const float* q = xyz2 + ((size_t)b * S + s) * 3;
        sx[s] = q[0]; sy[s] = q[1]; sz[s] = q[2];
    }
    __syncthreads();

    const float* p = xyz1 + ((size_t)b * N + n) * 3;
    const float px = p[0], py = p[1], pz = p[2];

    float d0 = 3.4e38f, d1 = 3.4e38f, d2 = 3.4e38f;
    int   i0 = 0, i1 = 0, i2 = 0;
    for (int s = 0; s < S; ++s) {
        float dx = px - sx[s], dy = py - sy[s], dz = pz - sz[s];
        float d = dx * dx + dy * dy + dz * dz;
        if (d < d0)      { d2 = d1; i2 = i1; d1 = d0; i1 = i0; d0 = d; i0 = s; }
        else if (d < d1) { d2 = d1; i2 = i1; d1 = d;  i1 = s; }
        else if (d < d2) { d2 = d;  i2 = s; }
    }
    float w0 = 1.f / (d0 + EPS_W);
    float w1 = 1.f / (d1 + EPS_W);
    float w2 = 1.f / (d2 + EPS_W);
    float inv = 1.f / (w0 + w1 + w2);

    const size_t o = ((size_t)b * N + n) * 3;
    w_ws[o + 0] = w0 * inv;  w_ws[o + 1] = w1 * inv;  w_ws[o + 2] = w2 * inv;
    idx_ws[o + 0] = i0;      idx_ws[o + 1] = i1;      idx_ws[o + 2] = i2;
}

// ---------------------------------------------------------------------------
// Kernel 2: GEMM (256xK @ KxN) + bias + BN + ReLU (+residual), per batch.
// Double-buffered LDS, async-to-LDS staging, 1 barrier per K-chunk.
// Layer 1 fuses the 3-NN interpolation into B-tile staging (software
// pipelined: gathers for chunk i+1 issued before compute(i), committed after).
// 8 waves, each a 64x32 tile = 4x2 grid of v_wmma_f32_16x16x4_f32 accums.
// ---------------------------------------------------------------------------
#define BM 256
#define BN 64
#define KC 16
#define A_STRIDE 20            // 80B rows: float4-aligned, conflict-free reads
#define B_STRIDE 80            // 320B rows: half-waves 32 banks apart
#define ABUF (BM * A_STRIDE)   // 5120 floats / buffer
#define BBUF (KC * B_STRIDE)   // 1280 floats / buffer

__global__ __launch_bounds__(256)
void gemm_bn_kernel(const float* __restrict__ W,      // (256,K) row-major
                    const float* __restrict__ X,      // (B,Kx,N); Kx=256 when fused
                    const float* __restrict__ P2,     // (B,512,S) points2 (fused)
                    const float* __restrict__ w_ws,   // (B,N,3)  (fused)
                    const int*   __restrict__ idx_ws, // (B,N,3)  (fused)
                    const float* __restrict__ bias,
                    const float* __restrict__ gamma,
                    const float* __restrict__ beta,
                    const float* __restrict__ mean,
                    const float* __restrict__ var,
                    const float* __restrict__ residual, // (B,256,N) or null
                    float* __restrict__ out,            // (B,256,N)
                    int K, int N, int S, int fuse)
{
    __shared__ float Alds[2 * ABUF];
    __shared__ float Blds[2 * BBUF];
    __shared__ float sw[BN * 3];
    __shared__ int   sidx[BN * 3];

    const int tid  = threadIdx.x;
    const int b    = blockIdx.y;
    const int n0   = blockIdx.x * BN;

    const int wv   = tid >> 5;
    const int lane = tid & 31;
    const int half = lane >> 4;         // lanes 16-31 take the K+2/K+3 pair
    const int ln   = lane & 15;
    const int wm   = (wv & 3) * 64;
    const int wn   = (wv >> 2) * 32;

    const int kl = tid & 15;            // B-staging chunk row
    const int nq = (tid >> 4) * 4;      // B-staging 4-wide column group

    const int Kx = fuse ? 256 : K;
    const float* Xb  = X + (size_t)b * Kx * N;
    const float* P2b = fuse ? (P2 + (size_t)b * 512 * S) : nullptr;

    // per-block interpolation weights/indices (depend only on n) -> LDS, once
    if (fuse) {
        for (int t = tid; t < BN * 3; t += 256) {
            const size_t o = ((size_t)b * N + n0 + t / 3) * 3 + (t % 3);
            sw[t]   = w_ws[o];
            sidx[t] = idx_ws[o];
        }
    }

    v8f acc[4][2];
#pragma unroll
    for (int mt = 0; mt < 4; ++mt)
#pragma unroll
        for (int nt = 0; nt < 2; ++nt)
            acc[mt][nt] = v8f{};

    const int nch = K / KC;
    float gv[4][3];                     // in-flight gather values (layer-1 rows)

    // ---- staging: issue phase (async DMAs + gather global loads) ----
    auto issue = [&](int ci, int bufsel) {
        const int kc = ci * KC;
        // A tile: W[tid][kc..kc+15] -> Alds row (async, 4x16B)
        const uint32_t aoff = lds_addr_of(&Alds[bufsel * ABUF + tid * A_STRIDE]);
        const uint32_t goff = ((uint32_t)tid * (uint32_t)K + (uint32_t)kc) * 4u;
#pragma unroll
        for (int j = 0; j < 4; ++j)
            async_load_b128(aoff + j * 16u, W, goff + j * 16u);
        // B tile
        const int kk = kc + kl;
        if (!fuse || kk < 256) {        // dense row: async straight to LDS
            const uint32_t boff =
                lds_addr_of(&Blds[bufsel * BBUF + kl * B_STRIDE + nq]);
            async_load_b128(boff, Xb,
                            ((uint32_t)kk * (uint32_t)N + (uint32_t)(n0 + nq)) * 4u);
        } else {                        // interp row: issue 12 L2 gathers
            const float* row = P2b + (size_t)(kk - 256) * S;
#pragma unroll
            for (int j = 0; j < 4; ++j) {
                const int lw = (nq + j) * 3;
                gv[j][0] = row[sidx[lw + 0]];
                gv[j][1] = row[sidx[lw + 1]];
                gv[j][2] = row[sidx[lw + 2]];
            }
        }
    };
    // ---- staging: commit phase for gather chunks (after compute overlap) ----
    auto commit_gather = [&](int bufsel) {
        float4 v;
        float t4[4];
#pragma unroll
        for (int j = 0; j < 4; ++j) {
            const int lw = (nq + j) * 3;
            t4[j] = sw[lw] * gv[j][0] + sw[lw + 1] * gv[j][1] + sw[lw + 2] * gv[j][2];
        }
        v = make_float4(t4[0], t4[1], t4[2], t4[3]);
        *reinterpret_cast<float4*>(&Blds[bufsel * BBUF + kl * B_STRIDE + nq]) = v;
    };
    auto is_gather = [&](int ci) { return fuse && (ci * KC >= 256); };

    // ---- prologue: stage chunk 0 into buffer 0 ----
    issue(0, 0);
    if (is_gather(0)) commit_gather(0);

    int buf = 0;
    for (int i = 0; i < nch; ++i) {
        wait_async0();                  // own chunk-i DMAs landed in LDS
        __syncthreads();                // everyone's chunk i staged; buf^1 free

        const bool has_next = (i + 1 < nch);
        if (has_next) issue(i + 1, buf ^ 1);   // overlap with compute below

        const float* Ab = &Alds[buf * ABUF];
        const float* Bb = &Blds[buf * BBUF];
#pragma unroll
        for (int ks = 0; ks < KC; ks += 4) {
            const int kb = ks + half * 2;
            v2f bfrag[2];
#pragma unroll
            for (int nt = 0; nt < 2; ++nt) {
                bfrag[nt].x = Bb[(kb + 0) * B_STRIDE + wn + nt * 16 + ln];
                bfrag[nt].y = Bb[(kb + 1) * B_STRIDE + wn + nt * 16 + ln];
            }
#pragma unroll
            for (int mt = 0; mt < 4; ++mt) {
                v2f afrag;
                afrag.x = Ab[(wm + mt * 16 + ln) * A_STRIDE + kb + 0];
                afrag.y = Ab[(wm + mt * 16 + ln) * A_STRIDE + kb + 1];
#pragma unroll
                for (int nt = 0; nt < 2; ++nt) {
                    acc[mt][nt] = __builtin_amdgcn_wmma_f32_16x16x4_f32(
                        false, afrag, false, bfrag[nt],
                        (short)0, acc[mt][nt], false, false);
                }
            }
        }

        if (has_next && is_gather(i + 1)) commit_gather(buf ^ 1);
        buf ^= 1;
    }

    // ---- epilogue: bias + BN fold + optional residual + ReLU ----
    const float* resb = residual ? residual + (size_t)b * 256 * N : nullptr;
    float* outb = out + (size_t)b * 256 * N;
#pragma unroll
    for (int mt = 0; mt < 4; ++mt) {
#pragma unroll
        for (int nt = 0; nt < 2; ++nt) {
            const int col = n0 + wn + nt * 16 + ln;
#pragma unroll
            for (int r = 0; r < 8; ++r) {
                const int row = wm + mt * 16 + r + half * 8;
                const float sc = gamma[row] * rsqrtf(var[row] + EPS_BN);
                float val = (acc[mt][nt][r] + bias[row] - mean[row]) * sc + beta[row];
                if (resb) val += resb[(size_t)row * N + col];
                val = fmaxf(val, 0.f);
                outb[(size_t)row * N + col] = val;
            }
        }
    }
}

// ---------------------------------------------------------------------------
// launch
// ---------------------------------------------------------------------------
extern "C" void kernel_launch(void* const* d_in, const int* in_sizes, int n_in,
                              void* d_out, int out_size, void* d_ws, size_t ws_size,
                              hipStream_t stream) {
    const float* xyz1    = (const float*)d_in[0];
    const float* xyz2    = (const float*)d_in[1];
    const float* points1 = (const float*)d_in[2];
    const float* points2 = (const float*)d_in[3];
    const float* W_fuse  = (const float*)d_in[4];
    const float* b_fuse  = (const float*)d_in[5];
    const float* g_fuse  = (const float*)d_in[6];
    const float* be_fuse = (const float*)d_in[7];
    const float* m_fuse  = (const float*)d_in[8];
    const float* v_fuse  = (const float*)d_in[9];
    const float* W1  = (const float*)d_in[10];
    const float* b1  = (const float*)d_in[11];
    const float* g1  = (const float*)d_in[12];
    const float* be1 = (const float*)d_in[13];
    const float* m1  = (const float*)d_in[14];
    const float* v1  = (const float*)d_in[15];
    const float* W2  = (const float*)d_in[16];
    const float* b2  = (const float*)d_in[17];
    const float* g2  = (const float*)d_in[18];
    const float* be2 = (const float*)d_in[19];
    const float* m2  = (const float*)d_in[20];
    const float* v2  = (const float*)d_in[21];

    const int B = 8, N = 4096, S = 1024;

    // workspace layout (~70 MB)
    float* w_ws   = (float*)d_ws;                         // B*N*3 floats
    int*   idx_ws = (int*)(w_ws + (size_t)B * N * 3);     // B*N*3 ints
    float* x1     = (float*)(idx_ws + (size_t)B * N * 3); // B*256*N
    float* h1     = x1 + (size_t)B * 256 * N;             // B*256*N

    knn3_kernel<<<dim3(N / 256, B), 256, 0, stream>>>(xyz1, xyz2, w_ws, idx_ws, N, S);

    const dim3 g(N / BN, B);
    // layer 1: fused concat(points1, interp(points2)) -> 256ch, BN, ReLU
    gemm_bn_kernel<<<g, 256, 0, stream>>>(W_fuse, points1, points2, w_ws, idx_ws,
                                          b_fuse, g_fuse, be_fuse, m_fuse, v_fuse,
                                          nullptr, x1, 768, N, S, 1);
    // layer 2: 256 -> 256, BN, ReLU
    gemm_bn_kernel<<<g, 256, 0, stream>>>(W1, x1, nullptr, nullptr, nullptr,
                                          b1, g1, be1, m1, v1,
                                          nullptr, h1, 256, N, S, 0);
    // layer 3: 256 -> 256, BN, +residual(x1), ReLU -> out
    gemm_bn_kernel<<<g, 256, 0, stream>>>(W2, h1, nullptr, nullptr, nullptr,
                                          b2, g2, be2, m2, v2,
                                          x1, (float*)d_out, 256, N, S, 0);
}